// SimpleGIN_80461917323503
// MI455X (gfx1250) — compile-verified
//
#include <hip/hip_runtime.h>

#define GIN_N 50000
#define GIN_D 128
#define GIN_E 800000
#define GIN_L 3
#define GIN_BN_EPS 1e-5f

typedef float v2f __attribute__((ext_vector_type(2)));
typedef float v8f __attribute__((ext_vector_type(8)));

// ---------------------------------------------------------------------------
// Kernel 1: agg[i] = (1 + eps[layer]) * h[i]      (vectorized float4 stream)
// ---------------------------------------------------------------------------
__global__ __launch_bounds__(256) void gin_init_agg(const float* __restrict__ h,
                                                    const float* __restrict__ eps,
                                                    int layer,
                                                    float* __restrict__ agg,
                                                    int total4) {
  int i = blockIdx.x * blockDim.x + threadIdx.x;
  if (i >= total4) return;
  const float s = 1.0f + eps[layer];
  float4 v = ((const float4*)h)[i];
  v.x *= s; v.y *= s; v.z *= s; v.w *= s;
  ((float4*)agg)[i] = v;
}

// ---------------------------------------------------------------------------
// Kernel 2: scatter-add  agg[dst] += h[src]  over edges.
// 32 lanes per edge, 4 floats per lane -> coalesced 512B row reads and
// native global_atomic_add_f32 (unsafeAtomicAdd avoids a CAS loop).
// This phase is the HBM roofline bound (~2.5 GB total across 3 layers).
// ---------------------------------------------------------------------------
__global__ __launch_bounds__(256) void gin_scatter_add(const float* __restrict__ h,
                                                       const int* __restrict__ edge,
                                                       float* __restrict__ agg) {
  int t = blockIdx.x * blockDim.x + threadIdx.x;
  int e = t >> 5;
  if (e >= GIN_E) return;
  int c = (t & 31) << 2;
  int src = edge[e];
  int dst = edge[GIN_E + e];
  const float4 v = *(const float4*)(h + (size_t)src * GIN_D + c);
  float* a = agg + (size_t)dst * GIN_D + c;
  unsafeAtomicAdd(a + 0, v.x);
  unsafeAtomicAdd(a + 1, v.y);
  unsafeAtomicAdd(a + 2, v.z);
  unsafeAtomicAdd(a + 3, v.w);
}

// ---------------------------------------------------------------------------
// Kernel 1b: pre-pack W matrices into WMMA f32 16x16x4 B-fragment order so the
// GEMM kernel issues one coalesced global_load_b64 per fragment.
// Packed index for matrix m: p = (kb*8 + nt)*32 + lane, value =
//   { W[(kb*4 + 2*(lane>>4) + 0)*128 + nt*16 + (lane&15)],
//     W[(kb*4 + 2*(lane>>4) + 1)*128 + nt*16 + (lane&15)] }
// blockIdx.x = matrix id 0..5  (id = 2*l + (0:W1, 1:W2))
// ---------------------------------------------------------------------------
__global__ __launch_bounds__(256) void gin_pack_w(const float* __restrict__ W1all,
                                                  const float* __restrict__ W2all,
                                                  v2f* __restrict__ pack) {
  const int id = blockIdx.x;          // 0..5
  const int l = id >> 1;
  const float* W = ((id & 1) ? W2all : W1all) + (size_t)l * GIN_D * GIN_D;
  v2f* P = pack + (size_t)id * 8192;  // 32 kb * 8 nt * 32 lanes
  for (int p = threadIdx.x; p < 8192; p += 256) {
    const int lane = p & 31;
    const int nt   = (p >> 5) & 7;
    const int kb   = p >> 8;
    const int bk   = (lane >> 4) * 2;
    const int n    = nt * 16 + (lane & 15);
    const int k0   = kb * 4 + bk;
    v2f v;
    v.x = W[(size_t)(k0 + 0) * GIN_D + n];
    v.y = W[(size_t)(k0 + 1) * GIN_D + n];
    P[p] = v;
  }
}

// ---------------------------------------------------------------------------
// Kernel 3: fused  z@W1 + b1 -> BN -> ReLU -> @W2 + b2 (-> ReLU)
// 128 threads = 4 waves; wave w computes TWO 16-row M-tiles (rows m0..m0+31),
// so every B fragment load feeds two v_wmma_f32_16x16x4_f32.
// The B stream is explicitly software-pipelined with a 4-slot register ring
// (depth-4 prefetch) so up to 4 global_load_b64 stay in flight instead of the
// serialized load->wait->wmma chain the scheduler produced under pressure.
// __launch_bounds__(128,1) relaxes the VGPR cap so the ring + 128 A-frag
// registers fit without spilling. EXEC stays all-ones around every WMMA.
// ---------------------------------------------------------------------------
__global__ __launch_bounds__(128, 1) void gin_mlp(const float* __restrict__ z,
                                                  const v2f* __restrict__ Wp1,
                                                  const v2f* __restrict__ Wp2,
                                                  const float* __restrict__ b1,
                                                  const float* __restrict__ gma,
                                                  const float* __restrict__ bta,
                                                  const float* __restrict__ mu,
                                                  const float* __restrict__ var_,
                                                  const float* __restrict__ b2,
                                                  float* __restrict__ out,
                                                  int relu_out) {
  __shared__ float mid[4][32][GIN_D];  // per-wave 32x128 intermediate (64 KB)

  const int tid  = threadIdx.x;
  const int wave = tid >> 5;
  const int lane = tid & 31;
  const int half = lane >> 4;   // 0: lanes 0-15, 1: lanes 16-31
  const int l15  = lane & 15;
  const int m0   = blockIdx.x * 128 + wave * 32;
  const int bk   = half * 2;    // K sub-offset within a k-block for A frags

  // Warm the phase-2 weight pack (64 KB) into cache while phase-1 math runs.
  // Lowers to global_prefetch_b8; no counters, no VGPR results.
#pragma unroll
  for (int i = 0; i < 4; ++i)
    __builtin_prefetch((const char*)Wp2 + (size_t)tid * 512 + (size_t)i * 128, 0, 1);

  // ---- preload A fragments for both M-tiles (clamped rows keep EXEC full;
  //      out-of-range rows are dropped at the final store) -------------------
  int rowLo = m0 + l15;       if (rowLo >= GIN_N) rowLo = GIN_N - 1;
  int rowHi = m0 + 16 + l15;  if (rowHi >= GIN_N) rowHi = GIN_N - 1;
  const float* apLo = z + (size_t)rowLo * GIN_D + bk;
  const float* apHi = z + (size_t)rowHi * GIN_D + bk;
  v2f aLo[32], aHi[32];
#pragma unroll
  for (int kb = 0; kb < 32; ++kb) {
    aLo[kb] = *(const v2f*)(apLo + kb * 4);
    aHi[kb] = *(const v2f*)(apHi + kb * 4);
  }

  // ---------------- GEMM1 + bias + BN + ReLU -> LDS ----------------
#pragma unroll
  for (int nt = 0; nt < 8; ++nt) {
    const int n = nt * 16 + l15;
    v8f accLo = {}, accHi = {};
    const v2f* bp = Wp1 + (size_t)nt * 32 + lane;
    v2f bring[4];                       // 4-deep B prefetch ring
#pragma unroll
    for (int i = 0; i < 4; ++i) bring[i] = bp[(size_t)i * 256];
#pragma unroll
    for (int kb = 0; kb < 32; ++kb) {
      const v2f bf = bring[kb & 3];
      if (kb + 4 < 32) bring[kb & 3] = bp[(size_t)(kb + 4) * 256];
      accLo = __builtin_amdgcn_wmma_f32_16x16x4_f32(false, aLo[kb], false, bf,
                                                    (short)0, accLo, false, false);
      accHi = __builtin_amdgcn_wmma_f32_16x16x4_f32(false, aHi[kb], false, bf,
                                                    (short)0, accHi, false, false);
    }
    const float scale = gma[n] * rsqrtf(var_[n] + GIN_BN_EPS);
    const float shift = (b1[n] - mu[n]) * scale + bta[n];
#pragma unroll
    for (int r = 0; r < 8; ++r) {
      mid[wave][r + half * 8][n]      = fmaxf(accLo[r] * scale + shift, 0.0f);
      mid[wave][16 + r + half * 8][n] = fmaxf(accHi[r] * scale + shift, 0.0f);
    }
  }

  __syncthreads();  // order LDS stores before the bulk A-fragment reload

  // ---- bulk-reload phase-2 A fragments from LDS (reuses the same regs) ----
#pragma unroll
  for (int kb = 0; kb < 32; ++kb) {
    aLo[kb] = *(const v2f*)&mid[wave][l15][kb * 4 + bk];
    aHi[kb] = *(const v2f*)&mid[wave][16 + l15][kb * 4 + bk];
  }

  // ---------------- GEMM2 + bias (+ReLU) -> global ----------------
#pragma unroll
  for (int nt = 0; nt < 8; ++nt) {
    const int n = nt * 16 + l15;
    v8f accLo = {}, accHi = {};
    const v2f* bp = Wp2 + (size_t)nt * 32 + lane;
    v2f bring[4];
#pragma unroll
    for (int i = 0; i < 4; ++i) bring[i] = bp[(size_t)i * 256];
#pragma unroll
    for (int kb = 0; kb < 32; ++kb) {
      const v2f bf = bring[kb & 3];
      if (kb + 4 < 32) bring[kb & 3] = bp[(size_t)(kb + 4) * 256];
      accLo = __builtin_amdgcn_wmma_f32_16x16x4_f32(false, aLo[kb], false, bf,
                                                    (short)0, accLo, false, false);
      accHi = __builtin_amdgcn_wmma_f32_16x16x4_f32(false, aHi[kb], false, bf,
                                                    (short)0, accHi, false, false);
    }
    const float bias = b2[n];
#pragma unroll
    for (int r = 0; r < 8; ++r) {
      float vLo = accLo[r] + bias;
      float vHi = accHi[r] + bias;
      if (relu_out) { vLo = fmaxf(vLo, 0.0f); vHi = fmaxf(vHi, 0.0f); }
      const int rLo = m0 + r + half * 8;
      const int rHi = m0 + 16 + r + half * 8;
      if (rLo < GIN_N) out[(size_t)rLo * GIN_D + n] = vLo;
      if (rHi < GIN_N) out[(size_t)rHi * GIN_D + n] = vHi;
    }
  }
}

// ---------------------------------------------------------------------------
// Host launcher. Inputs (setup_inputs order):
// 0:x 1:edge_index 2:W1 3:b1 4:gamma 5:beta 6:bn_mean 7:bn_var 8:W2 9:b2 10:eps
// d_ws layout: [0, N*D) agg | [N*D, 2*N*D) h ping | then 6*8192 v2f packed W.
// ---------------------------------------------------------------------------
extern "C" void kernel_launch(void* const* d_in, const int* in_sizes, int n_in,
                              void* d_out, int out_size, void* d_ws, size_t ws_size,
                              hipStream_t stream) {
  const float* x    = (const float*)d_in[0];
  const int*   edge = (const int*)d_in[1];
  const float* W1   = (const float*)d_in[2];
  const float* b1   = (const float*)d_in[3];
  const float* gma  = (const float*)d_in[4];
  const float* bta  = (const float*)d_in[5];
  const float* mu   = (const float*)d_in[6];
  const float* var_ = (const float*)d_in[7];
  const float* W2   = (const float*)d_in[8];
  const float* b2   = (const float*)d_in[9];
  const float* eps  = (const float*)d_in[10];
  float* out = (float*)d_out;

  float* agg  = (float*)d_ws;                      // N*D floats
  float* hbuf = agg + (size_t)GIN_N * GIN_D;       // N*D floats
  v2f*   wpk  = (v2f*)(hbuf + (size_t)GIN_N * GIN_D);  // 6 * 8192 v2f

  const int total4      = GIN_N * GIN_D / 4;
  const int init_blocks = (total4 + 255) / 256;
  const int scat_blocks = (GIN_E * 32 + 255) / 256;
  const int mlp_blocks  = (GIN_N + 127) / 128;

  // Repack all 6 weight matrices into B-fragment order (runs once per call;
  // deterministic, ~384 KB of work, negligible vs the main loop).
  gin_pack_w<<<2 * GIN_L, 256, 0, stream>>>(W1, W2, wpk);

  for (int l = 0; l < GIN_L; ++l) {
    const float* h  = (l == 0) ? x : hbuf;
    float*       hn = (l == GIN_L - 1) ? out : hbuf;  // MLP reads only agg,
                                                      // so in-place h is safe
    gin_init_agg<<<init_blocks, 256, 0, stream>>>(h, eps, l, agg, total4);
    gin_scatter_add<<<scat_blocks, 256, 0, stream>>>(h, edge, agg);
    gin_mlp<<<mlp_blocks, 128, 0, stream>>>(
        agg,
        wpk + (size_t)(2 * l + 0) * 8192,
        wpk + (size_t)(2 * l + 1) * 8192,
        b1 + (size_t)l * GIN_D,
        gma + (size_t)l * GIN_D, bta + (size_t)l * GIN_D,
        mu + (size_t)l * GIN_D, var_ + (size_t)l * GIN_D,
        b2 + (size_t)l * GIN_D,
        hn, (l < GIN_L - 1) ? 1 : 0);
  }
}